// DND_24438363914314
// MI455X (gfx1250) — compile-verified
//
#include <hip/hip_runtime.h>
#include <hip/hip_bf16.h>

// Problem constants (from reference): T=512, B=1024, DK=128, 2H=512, NH=4
#define TT   512
#define BB   1024
#define DKK  128
#define HH2  512        // 2*H
#define NHD  4

typedef __attribute__((ext_vector_type(2))) float v2f;
typedef __attribute__((ext_vector_type(8))) float v8f;

// ---------------------------------------------------------------------------
// Kernel A: fused per-batch attention read.
//   scores[m,n] = rpe[m,b] * dot(keys[m,b,:], query[b,n,:])   (m<min_step)
//   weight = softmax over m (per head n)
//   res[b, n*512+v] = sum_m weight[m,n] * vals[m,b,v]
// One block per batch b. 256 threads = 8 waves.
// ---------------------------------------------------------------------------
__global__ __launch_bounds__(256) void dnd_attn_kernel(
    const float* __restrict__ keys,    // [T,B,DK]
    const float* __restrict__ vals,    // [T,B,512]
    const float* __restrict__ rpe,     // [T,B,1]
    const float* __restrict__ query,   // [B,NH,DK]
    const int*   __restrict__ minstep, // scalar
    float*       __restrict__ res)     // [B, NH*512]
{
    __shared__ float s_q[NHD * DKK];   // 2 KB: query for this batch
    __shared__ float s_a[TT * NHD];    // 8 KB: scores -> softmax weights

    const int b    = blockIdx.x;
    const int t    = threadIdx.x;
    const int lane = t & 31;
    const int wave = t >> 5;
    const int m_steps = *minstep;      // uniform scalar load (device-side)

    // ---- load query[b] into LDS ----
    for (int i = t; i < NHD * DKK; i += 256)
        s_q[i] = query[(size_t)b * (NHD * DKK) + i];
    __syncthreads();

    // ---- stage 1: attention scores (each wave owns m = wave, wave+8, ...) ----
    for (int m = wave; m < m_steps; m += 8) {
        const float* kp = keys + ((size_t)m * BB + b) * DKK + lane * 4;
        float4 kv = *(const float4*)kp;                 // coalesced 512B/row
        float acc[NHD];
        #pragma unroll
        for (int n = 0; n < NHD; ++n) {
            const float* qp = s_q + n * DKK + lane * 4;
            acc[n] = kv.x * qp[0] + kv.y * qp[1] + kv.z * qp[2] + kv.w * qp[3];
        }
        #pragma unroll
        for (int off = 16; off > 0; off >>= 1) {
            #pragma unroll
            for (int n = 0; n < NHD; ++n)
                acc[n] += __shfl_xor(acc[n], off, 32);
        }
        if (lane == 0) {
            float r = rpe[(size_t)m * BB + b];          // RPE factors out of dot
            #pragma unroll
            for (int n = 0; n < NHD; ++n)
                s_a[m * NHD + n] = acc[n] * r;
        }
    }
    __syncthreads();

    // ---- stage 2: softmax over m per head (wave n in [0,4) owns head n) ----
    if (wave < NHD) {
        const int n = wave;
        float mx = -3.402823466e38f;
        for (int m = lane; m < m_steps; m += 32)
            mx = fmaxf(mx, s_a[m * NHD + n]);
        #pragma unroll
        for (int off = 16; off > 0; off >>= 1)
            mx = fmaxf(mx, __shfl_xor(mx, off, 32));
        float sum = 0.0f;
        for (int m = lane; m < m_steps; m += 32) {
            float e = __expf(s_a[m * NHD + n] - mx);
            s_a[m * NHD + n] = e;
            sum += e;
        }
        #pragma unroll
        for (int off = 16; off > 0; off >>= 1)
            sum += __shfl_xor(sum, off, 32);
        float inv = 1.0f / sum;
        for (int m = lane; m < m_steps; m += 32)
            s_a[m * NHD + n] *= inv;
    }
    __syncthreads();

    // ---- stage 3: weighted sum over memory (thread t owns v = 2t, 2t+1) ----
    float acc[NHD][2] = {};
    const int v0 = 2 * t;
    for (int m = 0; m < m_steps; ++m) {
        const float* vp = vals + ((size_t)m * BB + b) * HH2;
        if (m + 8 < m_steps)
            __builtin_prefetch(vp + 8 * (size_t)BB * HH2 + v0, 0, 1);
        float2 vv = *(const float2*)(vp + v0);          // coalesced 2KB/row
        float w0 = s_a[m * NHD + 0];                    // LDS broadcast
        float w1 = s_a[m * NHD + 1];
        float w2 = s_a[m * NHD + 2];
        float w3 = s_a[m * NHD + 3];
        acc[0][0] += w0 * vv.x;  acc[0][1] += w0 * vv.y;
        acc[1][0] += w1 * vv.x;  acc[1][1] += w1 * vv.y;
        acc[2][0] += w2 * vv.x;  acc[2][1] += w2 * vv.y;
        acc[3][0] += w3 * vv.x;  acc[3][1] += w3 * vv.y;
    }
    float* rp = res + (size_t)b * (NHD * HH2);
    #pragma unroll
    for (int n = 0; n < NHD; ++n) {
        float2 o; o.x = acc[n][0]; o.y = acc[n][1];
        *(float2*)(rp + n * HH2 + v0) = o;
    }
}

// ---------------------------------------------------------------------------
// Kernel B: out[B,512] = res[B,2048] @ W[512,2048]^T + bias,
// via v_wmma_f32_16x16x4_f32. Block = 128 threads = 4 waves; each wave owns
// one 16x16 output tile (same 16 batch rows, different 16 output cols).
// A-fragment (16x4 f32): lanes 0-15 -> M=lane, K={0,1}; lanes 16-31 -> K={2,3}
// B-fragment (4x16 f32): mirrored (lane%16 = N). Both are contiguous float2
// loads since res and W are K-major.
// C/D (16x16 f32): VGPR j: lanes 0-15 -> M=j, lanes 16-31 -> M=j+8; N=lane%16.
// ---------------------------------------------------------------------------
__global__ __launch_bounds__(128) void dnd_out_gemm(
    const float* __restrict__ res,   // [B, 2048]
    const float* __restrict__ Wm,    // [512, 2048]
    const float* __restrict__ bias,  // [512]
    float*       __restrict__ out)   // [B, 512]
{
    const int lane = threadIdx.x & 31;
    const int wave = threadIdx.x >> 5;
    const int col0 = blockIdx.x * 64 + wave * 16;   // output column tile
    const int row0 = blockIdx.y * 16;               // batch row tile
    const int mrow = lane & 15;
    const int koff = (lane < 16) ? 0 : 2;

    const float* ap = res + (size_t)(row0 + mrow) * 2048 + koff;
    const float* bp = Wm  + (size_t)(col0 + mrow) * 2048 + koff;

    v8f acc = {};
    #pragma unroll 4
    for (int k = 0; k < 2048; k += 4) {
        v2f a = *(const v2f*)(ap + k);              // b64 load: K pair
        v2f w = *(const v2f*)(bp + k);              // b64 load: K pair
        acc = __builtin_amdgcn_wmma_f32_16x16x4_f32(
            /*neg_a=*/false, a, /*neg_b=*/false, w,
            /*c_mod=*/(short)0, acc, /*reuse_a=*/false, /*reuse_b=*/false);
    }

    const float bv    = bias[col0 + mrow];
    const int   rbase = row0 + ((lane < 16) ? 0 : 8);
    #pragma unroll
    for (int j = 0; j < 8; ++j)
        out[(size_t)(rbase + j) * 512 + col0 + mrow] = acc[j] + bv;
}

// ---------------------------------------------------------------------------
extern "C" void kernel_launch(void* const* d_in, const int* in_sizes, int n_in,
                              void* d_out, int out_size, void* d_ws, size_t ws_size,
                              hipStream_t stream) {
    const float* keys    = (const float*)d_in[0];   // [T,B,DK]
    const float* vals    = (const float*)d_in[1];   // [T,B,512]
    const float* rpe     = (const float*)d_in[2];   // [T,B,1]
    const float* query   = (const float*)d_in[3];   // [B,NH,DK]
    const float* Wm      = (const float*)d_in[4];   // [512,2048]
    const float* bias    = (const float*)d_in[5];   // [512]
    const int*   minstep = (const int*)d_in[6];     // scalar
    float*       out     = (float*)d_out;           // [B,512]
    float*       res     = (float*)d_ws;            // [B,2048] = 8 MB scratch

    dnd_attn_kernel<<<BB, 256, 0, stream>>>(keys, vals, rpe, query, minstep, res);

    dim3 g2(HH2 / 64, BB / 16);                     // (8, 64)
    dnd_out_gemm<<<g2, 128, 0, stream>>>(res, Wm, bias, out);
}